// CReSTLoss_867583394249
// MI455X (gfx1250) — compile-verified
//
#include <hip/hip_runtime.h>
#include <stdint.h>

namespace {

constexpr int kC   = 1000;     // NUM_CLASSES
constexpr int kBx  = 8192;     // labeled batch
constexpr int kNu  = 65536;    // unlabeled batch
constexpr int kWavesPerBlock = 8;
constexpr int kBlock = kWavesPerBlock * 32;
constexpr int kChunk = 256;
constexpr int kNChunk = kNu / kChunk;  // 256
constexpr float kThresh = 0.95f;

typedef float v2f __attribute__((ext_vector_type(2)));
typedef float v8f __attribute__((ext_vector_type(8)));

#if defined(__has_builtin)
# if __has_builtin(__builtin_amdgcn_wmma_f32_16x16x4_f32)
#  define CREST_HAVE_WMMA4 1
# endif
#endif
#ifndef CREST_HAVE_WMMA4
# define CREST_HAVE_WMMA4 0
#endif

// ---- CDNA5 async global->LDS staging (ASYNCcnt path) ----------------------
__device__ __forceinline__ void async_ld_b128(uint32_t lds_byte_off, const void* gaddr) {
  // VDST = LDS byte offset VGPR, VADDR = 64-bit global address VGPR pair (GV mode)
  asm volatile("global_load_async_to_lds_b128 %0, %1, off"
               :: "v"(lds_byte_off), "v"(gaddr)
               : "memory");
}
__device__ __forceinline__ void wait_asynccnt0() {
  asm volatile("s_wait_asynccnt 0" ::: "memory");
}

// Stage one 1000-float row (4000 B, 16B aligned) into this wave's LDS slot.
__device__ __forceinline__ void stage_row(const float* __restrict__ grow,
                                          float* lrow, int lane) {
  // generic pointer into LDS: low 32 bits are the LDS byte offset
  uint32_t base = (uint32_t)(uintptr_t)(void*)lrow;
  const char* g = (const char*)grow;
  for (int ci = lane; ci < kC / 4; ci += 32)           // 250 b128 transfers
    async_ld_b128(base + (uint32_t)ci * 16u, g + (size_t)ci * 16u);
  wait_asynccnt0();                                     // per-wave counter
}

// ---- wave32 reductions ----------------------------------------------------
// Sum of the 32 per-lane values, via V_WMMA_F32_16X16X4_F32 with B == ones:
// A[m][0]=v_m, A[m][2]=v_{m+16} -> D[m][n] = v_m + v_{m+16}; per-lane sum of
// the 8 D VGPRs gives sum(m=0..7) on lanes 0-15 and sum(m=8..15) on 16-31.
__device__ __forceinline__ float wave_sum(float v) {
#if CREST_HAVE_WMMA4
  v2f a; a[0] = v;    a[1] = 0.0f;
  v2f b; b[0] = 1.0f; b[1] = 1.0f;
  v8f c = {};
  v8f d = __builtin_amdgcn_wmma_f32_16x16x4_f32(
      /*neg_a=*/false, a, /*neg_b=*/false, b,
      /*c_mod=*/(short)0, c, /*reuse_a=*/false, /*reuse_b=*/false);
  float s = d[0] + d[1] + d[2] + d[3] + d[4] + d[5] + d[6] + d[7];
  s += __shfl_xor(s, 16, 32);
  return s;
#else
  for (int off = 16; off > 0; off >>= 1) v += __shfl_xor(v, off, 32);
  return v;
#endif
}

__device__ __forceinline__ float wave_max(float v) {
  for (int off = 16; off > 0; off >>= 1) v = fmaxf(v, __shfl_xor(v, off, 32));
  return v;
}

// max + first-index argmax across the wave (smallest index on ties)
__device__ __forceinline__ void wave_max_arg(float& mv, int& mj) {
  for (int off = 16; off > 0; off >>= 1) {
    float ov = __shfl_xor(mv, off, 32);
    int   oj = __shfl_xor(mj, off, 32);
    if (ov > mv || (ov == mv && oj < mj)) { mv = ov; mj = oj; }
  }
}

// ---- kernel 1: per-row CE of logits_x -------------------------------------
__global__ void k_ce_x(const float* __restrict__ logits, const int* __restrict__ tgt,
                       float* __restrict__ ce_out) {
  __shared__ __align__(16) float lds[kWavesPerBlock][kC];
  const int wave = threadIdx.x >> 5, lane = threadIdx.x & 31;
  const int row = blockIdx.x * kWavesPerBlock + wave;
  if (row >= kBx) return;                    // wave-uniform
  float* l = lds[wave];
  stage_row(logits + (size_t)row * kC, l, lane);

  float mv = -3.402823466e38f;
  for (int j = lane; j < kC; j += 32) mv = fmaxf(mv, l[j]);
  mv = wave_max(mv);
  float s = 0.0f;
  for (int j = lane; j < kC; j += 32) s += expf(l[j] - mv);
  float tot = wave_sum(s);
  float lse = mv + logf(tot);
  if (lane == 0) ce_out[row] = lse - l[tgt[row]];
}

// ---- kernel 2: logits_wu -> max_probs, pseudo target t_m ------------------
__global__ void k_wu(const float* __restrict__ logits, float* __restrict__ maxp_out,
                     int* __restrict__ tm_out) {
  __shared__ __align__(16) float lds[kWavesPerBlock][kC];
  const int wave = threadIdx.x >> 5, lane = threadIdx.x & 31;
  const int row = blockIdx.x * kWavesPerBlock + wave;
  if (row >= kNu) return;
  float* l = lds[wave];
  stage_row(logits + (size_t)row * kC, l, lane);

  float mv = -3.402823466e38f; int mj = kC;
  for (int j = lane; j < kC; j += 32) {
    float v = l[j];
    if (v > mv) { mv = v; mj = j; }          // keeps first (smallest j) per lane
  }
  wave_max_arg(mv, mj);
  float s = 0.0f;
  for (int j = lane; j < kC; j += 32) s += expf(l[j] - mv);
  float tot = wave_sum(s);
  float lse = mv + logf(tot);
  float mp  = expf(mv - lse);                // max of softmax
  if (lane == 0) {
    maxp_out[row] = mp;
    tm_out[row]   = (mp >= kThresh) ? mj : 0;
  }
}

// ---- kernel 3: per-chunk class histograms ---------------------------------
__global__ void k_hist(const int* __restrict__ tm, int* __restrict__ chist) {
  __shared__ int h[kC];
  for (int k = threadIdx.x; k < kC; k += blockDim.x) h[k] = 0;
  __syncthreads();
  atomicAdd(&h[tm[blockIdx.x * kChunk + threadIdx.x]], 1);
  __syncthreads();
  for (int k = threadIdx.x; k < kC; k += blockDim.x)
    chist[(size_t)blockIdx.x * kC + k] = h[k];
}

// ---- kernel 4: serial per-class exclusive scan over chunks + bn -----------
__global__ void k_scan(int* __restrict__ chist, const float* __restrict__ gtp,
                       int* __restrict__ bn) {
  const int cls = blockIdx.x * blockDim.x + threadIdx.x;
  if (cls >= kC) return;
  int run = 0;
  for (int c = 0; c < kNChunk; ++c) {
    int t = chist[(size_t)c * kC + cls];
    chist[(size_t)c * kC + cls] = run;       // becomes exclusive prefix
    run += t;
  }
  // counts * gt_p, round half-to-even (matches jnp.round / torch.round)
  bn[cls] = (int)rintf((float)run * gtp[cls]);
}

// ---- kernel 5: index-ordered rank within class -> rebalance mask ----------
__global__ void k_rebal(const int* __restrict__ tm, const int* __restrict__ chist,
                        const int* __restrict__ bn, int* __restrict__ rebal) {
  __shared__ int stm[kChunk];
  const int i = blockIdx.x * kChunk + threadIdx.x;
  stm[threadIdx.x] = tm[i];
  __syncthreads();
  const int cls = stm[threadIdx.x];
  int rb = 0;
  if (cls != 0) {
    int local = 0;
    for (int k = 0; k < threadIdx.x; ++k) local += (stm[k] == cls);
    int rank = chist[(size_t)blockIdx.x * kC + cls] + local;
    rb = (rank < bn[cls]) ? 1 : 0;
  }
  rebal[i] = rb;
}

// ---- kernel 6: masked per-row CE of logits_su -----------------------------
__global__ void k_ce_su(const float* __restrict__ logits, const int* __restrict__ tm,
                        const int* __restrict__ rebal, float* __restrict__ ce_out) {
  __shared__ __align__(16) float lds[kWavesPerBlock][kC];
  const int wave = threadIdx.x >> 5, lane = threadIdx.x & 31;
  const int row = blockIdx.x * kWavesPerBlock + wave;
  if (row >= kNu) return;
  if (!rebal[row]) {                          // wave-uniform: skip HBM read
    if (lane == 0) ce_out[row] = 0.0f;
    return;
  }
  float* l = lds[wave];
  stage_row(logits + (size_t)row * kC, l, lane);

  float mv = -3.402823466e38f;
  for (int j = lane; j < kC; j += 32) mv = fmaxf(mv, l[j]);
  mv = wave_max(mv);
  float s = 0.0f;
  for (int j = lane; j < kC; j += 32) s += expf(l[j] - mv);
  float tot = wave_sum(s);
  float lse = mv + logf(tot);
  if (lane == 0) ce_out[row] = lse - l[tm[row]];
}

// ---- kernel 7: deterministic final reduction + scalar outputs -------------
__global__ void k_final(const float* __restrict__ cex, const float* __restrict__ cesu,
                        float* __restrict__ out) {
  __shared__ float p[256];
  const int tid = threadIdx.x;

  float s = 0.0f;
  for (int k = tid; k < kBx; k += 256) s += cex[k];
  p[tid] = s;
  __syncthreads();
  float lx = 0.0f;
  if (tid < 32) {
    float v = 0.0f;
    for (int k = 0; k < 8; ++k) v += p[tid + 32 * k];
    lx = wave_sum(v) / (float)kBx;
  }
  __syncthreads();

  s = 0.0f;
  for (int k = tid; k < kNu; k += 256) s += cesu[k];
  p[tid] = s;
  __syncthreads();
  float lu = 0.0f;
  if (tid < 32) {
    float v = 0.0f;
    for (int k = 0; k < 8; ++k) v += p[tid + 32 * k];
    lu = wave_sum(v) / (float)kNu;
  }
  if (tid == 0) {
    out[0] = lx + lu;   // LAMBDA_U == 1
    out[1] = lx;
    out[2] = lu;
  }
}

} // anonymous namespace

extern "C" void kernel_launch(void* const* d_in, const int* in_sizes, int n_in,
                              void* d_out, int out_size, void* d_ws, size_t ws_size,
                              hipStream_t stream) {
  (void)in_sizes; (void)n_in; (void)out_size; (void)ws_size;

  const float* logits_x  = (const float*)d_in[0];   // 8192 x 1000
  const float* logits_wu = (const float*)d_in[1];   // 65536 x 1000
  const float* logits_su = (const float*)d_in[2];   // 65536 x 1000
  const int*   targets_x = (const int*)d_in[3];     // 8192
  const float* gt_p      = (const float*)d_in[4];   // 1000
  // d_in[5] == t (unused)

  float* out = (float*)d_out;          // [loss, Lx, Lu, max_probs[65536]]

  // workspace layout (all 4B aligned), ~1.81 MB total
  char* ws = (char*)d_ws;
  float* ce_x   = (float*)(ws);                               //  8192 f32
  float* ce_su  = (float*)(ws + 32768);                       // 65536 f32
  int*   tm     = (int*)  (ws + 32768 + 262144);              // 65536 i32
  int*   rebal  = (int*)  (ws + 32768 + 262144 + 262144);     // 65536 i32
  int*   chist  = (int*)  (ws + 32768 + 262144 + 262144 + 262144);       // 256*1000 i32
  int*   bn     = (int*)  (ws + 32768 + 262144 + 262144 + 262144 + 1024000); // 1000 i32

  k_ce_x <<<kBx / kWavesPerBlock, kBlock, 0, stream>>>(logits_x, targets_x, ce_x);
  k_wu   <<<kNu / kWavesPerBlock, kBlock, 0, stream>>>(logits_wu, out + 3, tm);
  k_hist <<<kNChunk, kChunk, 0, stream>>>(tm, chist);
  k_scan <<<(kC + 255) / 256, 256, 0, stream>>>(chist, gt_p, bn);
  k_rebal<<<kNChunk, kChunk, 0, stream>>>(tm, chist, bn, rebal);
  k_ce_su<<<kNu / kWavesPerBlock, kBlock, 0, stream>>>(logits_su, tm, rebal, ce_su);
  k_final<<<1, 256, 0, stream>>>(ce_x, ce_su, out);
}